// GINSkipConnections_86423331930332
// MI455X (gfx1250) — compile-verified
//
#include <hip/hip_runtime.h>

#define N_NODES 100000
#define N_EDGES 1600000
#define N_GRAPHS 1000
#define DF 128
#define LATENT 64
#define BN_EPS 1e-5f

typedef float v2f __attribute__((ext_vector_type(2)));
typedef float v8f __attribute__((ext_vector_type(8)));

__device__ __forceinline__ float leaky(float v) { return v >= 0.f ? v : 0.2f * v; }

// CDNA5 async DMA: global -> LDS, 16 bytes per lane, tracked by ASYNCcnt.
__device__ __forceinline__ void async_copy16(const float* gptr, float* lptr) {
    unsigned lds = (unsigned)(size_t)lptr;              // low 32b of shared aperture = LDS offset
    unsigned long long ga = (unsigned long long)gptr;   // global address
    asm volatile("global_load_async_to_lds_b128 %0, %1, off"
                 :
                 : "v"(lds), "v"(ga)
                 : "memory");
}
__device__ __forceinline__ void wait_async0() {
    asm volatile("s_wait_asynccnt 0" ::: "memory");
}

// ---------------------------------------------------------------- utilities
__global__ void zero_kernel(float* __restrict__ p, int n) {
    int i = blockIdx.x * blockDim.x + threadIdx.x;
    if (i < n) p[i] = 0.f;
}

// ------------------------------------------------- edge scatter: agg[dst] += x[src]
__global__ __launch_bounds__(256) void edge_scatter_kernel(const float* __restrict__ x,
                                                           const int* __restrict__ ei,
                                                           float* __restrict__ agg) {
    int wave = threadIdx.x >> 5;
    int lane = threadIdx.x & 31;
    long e = (long)blockIdx.x * 8 + wave;
    if (e >= N_EDGES) return;
    int src = ei[e];
    int dst = ei[(long)N_EDGES + e];
    const float4 v = *(const float4*)(x + (long)src * DF + lane * 4);
    float* ap = agg + (long)dst * DF + lane * 4;
    atomicAdd(ap + 0, v.x);
    atomicAdd(ap + 1, v.y);
    atomicAdd(ap + 2, v.z);
    atomicAdd(ap + 3, v.w);
}

// ------------------------------------------------- GEMM1: H = leaky(A @ W + b)
// block = 256 threads = 8 waves; block owns 16 rows; wave w owns cols [16w,16w+16)
__global__ __launch_bounds__(256) void gemm1_kernel(const float* __restrict__ A,
                                                    const float* __restrict__ W,
                                                    const float* __restrict__ bias,
                                                    float* __restrict__ H) {
    __shared__ float As[16 * 132];
    const int row0 = blockIdx.x * 16;
    const int t = threadIdx.x;
#pragma unroll
    for (int i = 0; i < 2; ++i) {
        int idx = t * 4 + i * 1024;
        int r = idx >> 7, c = idx & 127;
        async_copy16(A + (long)(row0 + r) * DF + c, As + r * 132 + c);
    }
    wait_async0();
    __syncthreads();

    const int wave = t >> 5, lane = t & 31;
    const int col0 = wave * 16;
    const int mn = lane & 15;    // A row within tile / B col within tile
    const int half = lane >> 4;  // selects K pair {0,1} vs {2,3}
    v8f acc = {};
    for (int k = 0; k < DF; k += 4) {
        int ko = k + half * 2;
        v2f a = *(const v2f*)(As + mn * 132 + ko);
        const float* wp = W + (long)ko * DF + col0 + mn;
        v2f b;
        b.x = wp[0];
        b.y = wp[DF];
        acc = __builtin_amdgcn_wmma_f32_16x16x4_f32(false, a, false, b, (short)0, acc,
                                                    false, false);
    }
    const int col = col0 + mn;
    const float bv = bias[col];
    const int rbase = row0 + half * 8;
#pragma unroll
    for (int i = 0; i < 8; ++i) {
        H[(long)(rbase + i) * DF + col] = leaky(acc[i] + bv);
    }
}

// ------------------------------------------------- column stats (sum, sumsq) w/ atomics
__global__ __launch_bounds__(128) void colstats_kernel(const float* __restrict__ H,
                                                       float* __restrict__ stats) {
    int c = threadIdx.x;
    int r0 = blockIdx.x * 256;
    int rend = r0 + 256;
    if (rend > N_NODES) rend = N_NODES;
    float s = 0.f, s2 = 0.f;
    for (int r = r0; r < rend; ++r) {
        float v = H[(long)r * DF + c];
        s += v;
        s2 += v * v;
    }
    atomicAdd(&stats[c], s);
    atomicAdd(&stats[DF + c], s2);
}

// ------------------------------------------------- fold stats into scale/shift
__global__ __launch_bounds__(128) void bnfinal_kernel(const float* __restrict__ g,
                                                      const float* __restrict__ bt,
                                                      float* __restrict__ stats,
                                                      float invN) {
    int c = threadIdx.x;
    float mean = stats[c] * invN;
    float var = stats[DF + c] * invN - mean * mean;
    float sc = g[c] * rsqrtf(var + BN_EPS);
    stats[2 * DF + c] = sc;
    stats[3 * DF + c] = bt[c] - mean * sc;
}

// ------------------------------------------------- fold BN affine into W2/b2:
// BN(h) @ W2 + b2 == h @ (diag(sc) W2) + (sh @ W2 + b2)
__global__ __launch_bounds__(128) void foldw2_kernel(const float* __restrict__ W2,
                                                     const float* __restrict__ b2,
                                                     const float* __restrict__ stats,
                                                     float* __restrict__ W2p,
                                                     float* __restrict__ b2p) {
    int n = threadIdx.x;
    float acc = b2[n];
    for (int k = 0; k < DF; ++k) {
        float w = W2[(long)k * DF + n];
        acc += stats[3 * DF + k] * w;
        W2p[(long)k * DF + n] = stats[2 * DF + k] * w;
    }
    b2p[n] = acc;
}

// ------------------------------------------------- GEMM2: X = leaky(H @ W2' + b2') + X
__global__ __launch_bounds__(256) void gemm2_kernel(const float* __restrict__ H,
                                                    const float* __restrict__ W,
                                                    const float* __restrict__ bias,
                                                    float* __restrict__ X) {
    __shared__ float As[16 * 132];
    const int row0 = blockIdx.x * 16;
    const int t = threadIdx.x;
#pragma unroll
    for (int i = 0; i < 2; ++i) {
        int idx = t * 4 + i * 1024;
        int r = idx >> 7, c = idx & 127;
        async_copy16(H + (long)(row0 + r) * DF + c, As + r * 132 + c);
    }
    wait_async0();
    __syncthreads();

    const int wave = t >> 5, lane = t & 31;
    const int col0 = wave * 16;
    const int mn = lane & 15;
    const int half = lane >> 4;
    v8f acc = {};
    for (int k = 0; k < DF; k += 4) {
        int ko = k + half * 2;
        v2f a = *(const v2f*)(As + mn * 132 + ko);
        const float* wp = W + (long)ko * DF + col0 + mn;
        v2f b;
        b.x = wp[0];
        b.y = wp[DF];
        acc = __builtin_amdgcn_wmma_f32_16x16x4_f32(false, a, false, b, (short)0, acc,
                                                    false, false);
    }
    const int col = col0 + mn;
    const float bv = bias[col];
    const int rbase = row0 + half * 8;
#pragma unroll
    for (int i = 0; i < 8; ++i) {
        long idx = (long)(rbase + i) * DF + col;
        X[idx] = leaky(acc[i] + bv) + X[idx];
    }
}

// ------------------------------------------------- graph pooling via atomics
__global__ __launch_bounds__(256) void pool_kernel(const float* __restrict__ X,
                                                   const int* __restrict__ batch,
                                                   float* __restrict__ pooled) {
    int wave = threadIdx.x >> 5;
    int lane = threadIdx.x & 31;
    long n = (long)blockIdx.x * 8 + wave;
    if (n >= N_NODES) return;
    int g = batch[n];
    const float4 v = *(const float4*)(X + n * DF + lane * 4);
    float* pp = pooled + (long)g * DF + lane * 4;
    atomicAdd(pp + 0, v.x);
    atomicAdd(pp + 1, v.y);
    atomicAdd(pp + 2, v.z);
    atomicAdd(pp + 3, v.w);
}

// ------------------------------------------------- pooled BN stats (single block)
__global__ __launch_bounds__(128) void pooledbn_kernel(const float* __restrict__ P,
                                                       const float* __restrict__ g,
                                                       const float* __restrict__ bt,
                                                       float* __restrict__ pstats) {
    int c = threadIdx.x;
    float s = 0.f, s2 = 0.f;
    for (int r = 0; r < N_GRAPHS; ++r) {
        float v = P[(long)r * DF + c];
        s += v;
        s2 += v * v;
    }
    float invN = 1.0f / (float)N_GRAPHS;
    float mean = s * invN;
    float var = s2 * invN - mean * mean;
    float sc = g[c] * rsqrtf(var + BN_EPS);
    pstats[c] = sc;
    pstats[DF + c] = bt[c] - mean * sc;
}

// ------------------------------------------------- fold pooled BN into fc weights
__global__ __launch_bounds__(64) void foldfc_kernel(const float* __restrict__ fcW,
                                                    const float* __restrict__ fcb,
                                                    const float* __restrict__ pstats,
                                                    float* __restrict__ fcWp,
                                                    float* __restrict__ fcbp) {
    int n = threadIdx.x;
    float acc = fcb[n];
    for (int k = 0; k < DF; ++k) {
        float w = fcW[(long)k * LATENT + n];
        acc += pstats[DF + k] * w;
        fcWp[(long)k * LATENT + n] = pstats[k] * w;
    }
    fcbp[n] = acc;
}

// ------------------------------------------------- final: out = pooled @ fcW' + fcb'
__global__ __launch_bounds__(128) void gemmfc_kernel(const float* __restrict__ P,
                                                     const float* __restrict__ W,
                                                     const float* __restrict__ bias,
                                                     float* __restrict__ out) {
    __shared__ float As[16 * 132];
    const int row0 = blockIdx.x * 16;
    const int t = threadIdx.x;
#pragma unroll
    for (int i = 0; i < 4; ++i) {
        int idx = t * 4 + i * 512;
        int r = idx >> 7, c = idx & 127;
        int gr = row0 + r;
        if (gr < N_GRAPHS) {
            async_copy16(P + (long)gr * DF + c, As + r * 132 + c);
        } else {
            *(float4*)(As + r * 132 + c) = make_float4(0.f, 0.f, 0.f, 0.f);
        }
    }
    wait_async0();
    __syncthreads();

    const int wave = t >> 5, lane = t & 31;
    const int col0 = wave * 16;
    const int mn = lane & 15;
    const int half = lane >> 4;
    v8f acc = {};
    for (int k = 0; k < DF; k += 4) {
        int ko = k + half * 2;
        v2f a = *(const v2f*)(As + mn * 132 + ko);
        const float* wp = W + (long)ko * LATENT + col0 + mn;
        v2f b;
        b.x = wp[0];
        b.y = wp[LATENT];
        acc = __builtin_amdgcn_wmma_f32_16x16x4_f32(false, a, false, b, (short)0, acc,
                                                    false, false);
    }
    const int col = col0 + mn;
    const float bv = bias[col];
    const int rbase = row0 + half * 8;
#pragma unroll
    for (int i = 0; i < 8; ++i) {
        int r = rbase + i;
        if (r < N_GRAPHS) out[(long)r * LATENT + col] = acc[i] + bv;
    }
}

extern "C" void kernel_launch(void* const* d_in, const int* in_sizes, int n_in,
                              void* d_out, int out_size, void* d_ws, size_t ws_size,
                              hipStream_t stream) {
    const float* x    = (const float*)d_in[0];
    const int*   ei   = (const int*)d_in[1];
    const int*   batch= (const int*)d_in[2];
    const float* W1   = (const float*)d_in[3];
    const float* b1   = (const float*)d_in[4];
    const float* g1   = (const float*)d_in[5];
    const float* bt1  = (const float*)d_in[6];
    const float* W2   = (const float*)d_in[7];
    const float* b2   = (const float*)d_in[8];
    const float* bn_g = (const float*)d_in[9];
    const float* bn_b = (const float*)d_in[10];
    const float* fc_W = (const float*)d_in[11];
    const float* fc_b = (const float*)d_in[12];
    float* out = (float*)d_out;

    const size_t NF = (size_t)N_NODES * DF;
    float* x_ws   = (float*)d_ws;
    float* agg    = x_ws + NF;
    float* h      = agg + NF;
    float* stats  = h + NF;                          // 4*DF
    float* pooled = stats + 4 * DF;                  // N_GRAPHS*DF
    float* pstats = pooled + (size_t)N_GRAPHS * DF;  // 2*DF
    float* W2p    = pstats + 2 * DF;                 // DF*DF
    float* b2p    = W2p + (size_t)DF * DF;           // DF
    float* fcWp   = b2p + DF;                        // DF*LATENT
    float* fcbp   = fcWp + (size_t)DF * LATENT;      // LATENT

    hipMemcpyAsync(x_ws, x, NF * sizeof(float), hipMemcpyDeviceToDevice, stream);

    const int ROW_TILES = N_NODES / 16;  // 6250, exact
    for (int l = 0; l < 3; ++l) {
        hipMemcpyAsync(agg, x_ws, NF * sizeof(float), hipMemcpyDeviceToDevice, stream);
        edge_scatter_kernel<<<N_EDGES / 8, 256, 0, stream>>>(x_ws, ei, agg);
        zero_kernel<<<2, 256, 0, stream>>>(stats, 4 * DF);
        gemm1_kernel<<<ROW_TILES, 256, 0, stream>>>(agg, W1 + (size_t)l * DF * DF,
                                                    b1 + l * DF, h);
        colstats_kernel<<<(N_NODES + 255) / 256, 128, 0, stream>>>(h, stats);
        bnfinal_kernel<<<1, 128, 0, stream>>>(g1 + l * DF, bt1 + l * DF, stats,
                                              1.0f / (float)N_NODES);
        foldw2_kernel<<<1, 128, 0, stream>>>(W2 + (size_t)l * DF * DF, b2 + l * DF,
                                             stats, W2p, b2p);
        gemm2_kernel<<<ROW_TILES, 256, 0, stream>>>(h, W2p, b2p, x_ws);
    }

    zero_kernel<<<(N_GRAPHS * DF + 255) / 256, 256, 0, stream>>>(pooled, N_GRAPHS * DF);
    pool_kernel<<<(N_NODES + 7) / 8, 256, 0, stream>>>(x_ws, batch, pooled);
    pooledbn_kernel<<<1, 128, 0, stream>>>(pooled, bn_g, bn_b, pstats);
    foldfc_kernel<<<1, 64, 0, stream>>>(fc_W, fc_b, pstats, fcWp, fcbp);
    gemmfc_kernel<<<(N_GRAPHS + 15) / 16, 128, 0, stream>>>(pooled, fcWp, fcbp, out);
}